// GaussianVAE_79396765434420
// MI455X (gfx1250) — compile-verified
//
#include <hip/hip_runtime.h>
#include <hip/hip_bf16.h>

// Problem constants from the reference: B=4, N=8192, P=2, G=512.
#define NN_B 4
#define NN_N 8192
#define NN_G 512

typedef float v2f __attribute__((ext_vector_type(2)));
typedef float v8f __attribute__((ext_vector_type(8)));

// One wave (32 threads) per block of 16 predicted points.
// A-matrix (16x4 f32): lanes 0-15 carry K=0,1 (=px,py) for M=lane; lanes 16-31
// carry K=2,3 which are zero (P=2 padded to K=4).
// B-matrix (4x16 f32): lanes 0-15 carry K=0,1 (=rx,ry) for N=lane; lanes 16-31 zero.
// D tile (16x16 f32): VGPR v holds row M=v (lanes 0-15, col N=lane) and row
// M=v+8 (lanes 16-31, col N=lane-16).
__global__ __launch_bounds__(32) void nn_argmin_kernel(
    const float* __restrict__ pred, const float* __restrict__ realp,
    int* __restrict__ nearest)
{
    const int lane  = threadIdx.x;      // 0..31 (wave32)
    const int l15   = lane & 15;
    const int i0    = blockIdx.x * 16;  // predicted-row block
    const int b     = blockIdx.y;       // batch
    const bool loHalf = lane < 16;

    const float2* pp = (const float2*)pred  + (size_t)b * NN_N + i0;
    const float2* rp = (const float2*)realp + (size_t)b * NN_N;

    // Load A once (zero the K=2,3 half).
    float2 av = pp[l15];
    v2f A;
    A.x = loHalf ? av.x : 0.0f;
    A.y = loHalf ? av.y : 0.0f;

    float bestD[8];
    int   bestIdx[8];
#pragma unroll
    for (int v = 0; v < 8; ++v) { bestD[v] = __builtin_inff(); bestIdx[v] = 0; }

    // Software-pipeline the 16-point real tile one iteration ahead.
    float2 r = rp[l15];
    for (int j0 = 0; j0 < NN_N; j0 += 16) {
        const int jn = (j0 + 16) & (NN_N - 1);   // wraps harmlessly on last iter
        float2 rnext = rp[jn + l15];

        // rn needed by ALL lanes (each lane owns column j = j0 + (lane&15)).
        float rn = __builtin_fmaf(r.y, r.y, r.x * r.x);

        // B operand: zero lanes 16-31 (K=2,3 rows of the 4x16 matrix).
        v2f Bv;
        Bv.x = loHalf ? r.x : 0.0f;
        Bv.y = loHalf ? r.y : 0.0f;

        v8f cz = {};
        // D = A x B  (cross terms p·r for a 16x16 tile), C = 0.
        v8f d = __builtin_amdgcn_wmma_f32_16x16x4_f32(
            /*neg_a=*/false, A, /*neg_b=*/false, Bv,
            /*c_mod=*/(short)0, cz, /*reuse_a=*/false, /*reuse_b=*/false);

        const int jl = j0 + l15;
#pragma unroll
        for (int v = 0; v < 8; ++v) {
            // pn[i] is constant along j, so argmin key = rn - 2*cross.
            float key = __builtin_fmaf(d[v], -2.0f, rn);
            bool  lt  = key < bestD[v];      // strict < keeps earliest j (ties)
            bestD[v]   = lt ? key : bestD[v];
            bestIdx[v] = lt ? jl  : bestIdx[v];
        }
        r = rnext;
    }

    // Cross-lane argmin within each 16-lane half (xor 1,2,4,8 stays in-half).
#pragma unroll
    for (int v = 0; v < 8; ++v) {
        float dmin = bestD[v];
        int   idx  = bestIdx[v];
#pragma unroll
        for (int m = 1; m < 16; m <<= 1) {
            float od = __shfl_xor(dmin, m, 32);
            int   oi = __shfl_xor(idx,  m, 32);
            bool take = (od < dmin) || ((od == dmin) && (oi < idx));
            dmin = take ? od : dmin;
            idx  = take ? oi : idx;
        }
        if (lane == 0)
            nearest[(size_t)b * NN_N + i0 + v] = idx;
        if (lane == 16)
            nearest[(size_t)b * NN_N + i0 + 8 + v] = idx;
    }
}

// Bandwidth phase: out[b,i,:] = expr[b, nearest[b,i], :].
// One float4 per thread, 2 rows (2*512 floats) per 256-thread block.
__global__ __launch_bounds__(256) void nn_gather_kernel(
    const float* __restrict__ expr, const int* __restrict__ nearest,
    float* __restrict__ out)
{
    const int tid = threadIdx.x;
    const int row = blockIdx.x * 2 + (tid >> 7);   // global (b*N + i)
    const int col = tid & 127;                     // float4 column, G/4 = 128
    const int b   = row >> 13;                     // N = 8192 = 2^13
    const int j   = nearest[row];

    const float4* src = (const float4*)(expr + ((size_t)b * NN_N + j) * NN_G);
    float4*       dst = (float4*)(out + (size_t)row * NN_G);
    dst[col] = src[col];
}

extern "C" void kernel_launch(void* const* d_in, const int* in_sizes, int n_in,
                              void* d_out, int out_size, void* d_ws, size_t ws_size,
                              hipStream_t stream) {
    const float* pred  = (const float*)d_in[0];  // (B, N, 2) f32
    const float* realp = (const float*)d_in[1];  // (B, N, 2) f32
    const float* expr  = (const float*)d_in[2];  // (B, N, G) f32
    float* out = (float*)d_out;                  // (B, N, G) f32
    int* nearest = (int*)d_ws;                   // B*N ints = 128 KB scratch

    dim3 grid1(NN_N / 16, NN_B);                 // 512 x 4 waves
    nn_argmin_kernel<<<grid1, 32, 0, stream>>>(pred, realp, nearest);

    dim3 grid2((NN_B * NN_N) / 2);               // 2 rows per block
    nn_gather_kernel<<<grid2, 256, 0, stream>>>(expr, nearest, out);
}